// GAT_80376017977912
// MI455X (gfx1250) — compile-verified
//
#include <hip/hip_runtime.h>
#include <hip/hip_bf16.h>
#include <math.h>

typedef __attribute__((ext_vector_type(2))) float v2f;
typedef __attribute__((ext_vector_type(8))) float v8f;

#define NEG_SLOPE 0.2f

// ---------------------------------------------------------------------------
// WMMA GEMM (fp32): Hout[N x NOUT] = X[N x KDIM] * W[KDIM x NOUT]
// Uses V_WMMA_F32_16X16X4_F32: A 16x4, B 4x16, C/D 16x16, all f32.
// Each wave computes one 16x16 output tile; CT waves per block share the
// LDS-staged 16-row A strip and the transposed B column block.
// Row pitch padded by 2 floats: 520B pitch avoids single-bank hotspots and
// keeps 8B alignment for ds_load_b64 pair reads.
// ---------------------------------------------------------------------------
template<int KDIM, int CT>
__global__ __launch_bounds__(CT * 32)
void gemm_wmma_kernel(const float* __restrict__ X, const float* __restrict__ W,
                      float* __restrict__ Hout, int N, int NOUT) {
    __shared__ float sA[16][KDIM + 2];        // 16 x K strip of X
    __shared__ float sBt[CT * 16][KDIM + 2];  // [n][k] transposed W block

    const int tid  = threadIdx.x;
    const int lane = tid & 31;
    const int wave = tid >> 5;
    const int row0 = blockIdx.x * 16;
    const int colBlock = blockIdx.y * CT * 16;
    const int col0 = colBlock + wave * 16;

    // Stage A strip with 8-byte vector copies (pure copy, no conversion).
    for (int i = tid; i < 16 * (KDIM / 2); i += CT * 32) {
        int r = i / (KDIM / 2);
        int c = (i % (KDIM / 2)) * 2;
        int rg = row0 + r; if (rg >= N) rg = N - 1;
        *(v2f*)&sA[r][c] = *(const v2f*)&X[(size_t)rg * KDIM + c];
    }
    // Stage transposed W block: sBt[n][k] = W[k][colBlock+n].
    // Consecutive threads read consecutive n -> coalesced global reads.
    for (int i = tid; i < KDIM * CT * 16; i += CT * 32) {
        int n = i % (CT * 16);
        int k = i / (CT * 16);
        sBt[n][k] = W[(size_t)k * NOUT + colBlock + n];
    }
    __syncthreads();

    v8f acc = {};
    const int mn = lane & 15;
    const int kb = (lane < 16) ? 0 : 2;   // 32-bit A/B layout: lane half selects K pair

    const float* arow = &sA[mn][0];
    const float* brow = &sBt[wave * 16 + mn][0];
#pragma unroll
    for (int kt = 0; kt < KDIM / 4; ++kt) {
        v2f a = *(const v2f*)&arow[kt * 4 + kb];
        v2f b = *(const v2f*)&brow[kt * 4 + kb];
        acc = __builtin_amdgcn_wmma_f32_16x16x4_f32(
            false, a, false, b, (short)0, acc, false, false);
    }

    // C/D layout: VGPR g -> M = g (+8 for lanes 16..31), N = lane%16
    const int rbase = (lane < 16) ? 0 : 8;
#pragma unroll
    for (int g = 0; g < 8; ++g) {
        int rg = row0 + rbase + g;
        if (rg < N) Hout[(size_t)rg * NOUT + col0 + mn] = acc[g];
    }
}

// ---------------------------------------------------------------------------
// Per-node attention terms: a_src[n,h] = sum_o h[n,h,o]*att_s[h,o]
// ---------------------------------------------------------------------------
template<int H, int O>
__global__ void att_coeff_kernel(const float* __restrict__ h,
                                 const float* __restrict__ att_s,
                                 const float* __restrict__ att_d,
                                 float* __restrict__ as, float* __restrict__ ad,
                                 int N) {
    long total = (long)N * H;
    long stride = (long)gridDim.x * blockDim.x;
    for (long i = (long)blockIdx.x * blockDim.x + threadIdx.x; i < total; i += stride) {
        int n = (int)(i / H), hh = (int)(i % H);
        const float* hp = h + (size_t)n * (H * O) + hh * O;
        float s = 0.f, d = 0.f;
#pragma unroll
        for (int o = 0; o < O; ++o) {
            float v = hp[o];
            s += v * att_s[hh * O + o];
            d += v * att_d[hh * O + o];
        }
        as[i] = s; ad[i] = d;
    }
}

__global__ void zero_kernel(float* __restrict__ p, long n) {
    long stride = (long)gridDim.x * blockDim.x;
    for (long i = (long)blockIdx.x * blockDim.x + threadIdx.x; i < n; i += stride)
        p[i] = 0.f;
}

// denom[d,h] += exp(leaky_relu(a_src[s,h] + a_dst[d,h]))
template<int H>
__global__ void edge_denom_kernel(const int* __restrict__ src, const int* __restrict__ dst,
                                  int E, int N,
                                  const float* __restrict__ as, const float* __restrict__ ad,
                                  float* __restrict__ denom) {
    long total = (long)E + N;
    long stride = (long)gridDim.x * blockDim.x;
    for (long e = (long)blockIdx.x * blockDim.x + threadIdx.x; e < total; e += stride) {
        int s, d;
        if (e < E) { s = src[e]; d = dst[e]; } else { s = d = (int)(e - E); }
#pragma unroll
        for (int hh = 0; hh < H; ++hh) {
            float l = as[(size_t)s * H + hh] + ad[(size_t)d * H + hh];
            l = (l > 0.f) ? l : NEG_SLOPE * l;
            atomicAdd(&denom[(size_t)d * H + hh], __expf(l));
        }
    }
}

// agg[d, h, :] += h[s, h, :] * exp(leaky(..)) / denom[d,h]   (one wave per edge)
template<int H, int O>
__global__ void edge_aggregate_kernel(const int* __restrict__ src, const int* __restrict__ dst,
                                      int E, int N,
                                      const float* __restrict__ as, const float* __restrict__ ad,
                                      const float* __restrict__ denom,
                                      const float* __restrict__ h,
                                      float* __restrict__ agg) {
    const int HO = H * O;
    const int lane = threadIdx.x & 31;
    long wid = ((long)blockIdx.x * blockDim.x + threadIdx.x) >> 5;
    long nw  = ((long)gridDim.x * blockDim.x) >> 5;
    long total = (long)E + N;
    for (long e = wid; e < total; e += nw) {
        int s, d;
        if (e < E) { s = src[e]; d = dst[e]; } else { s = d = (int)(e - E); }

        // Prefetch the next edge's source feature row (L2-resident gather).
        long en = e + nw;
        if (en < total) {
            int sn = (en < E) ? src[en] : (int)(en - E);
            if (lane < HO)
                __builtin_prefetch(&h[(size_t)sn * HO + lane], 0, 0);
        }

        float w[H];
#pragma unroll
        for (int hh = 0; hh < H; ++hh) {
            float l = as[(size_t)s * H + hh] + ad[(size_t)d * H + hh];
            l = (l > 0.f) ? l : NEG_SLOPE * l;
            w[hh] = __expf(l) / denom[(size_t)d * H + hh];
        }
        for (int j = lane; j < HO; j += 32) {
            int hh = j / O;
            atomicAdd(&agg[(size_t)d * HO + j], h[(size_t)s * HO + j] * w[hh]);
        }
    }
}

// out = elu(agg + b)   (in place allowed)
__global__ void bias_elu_kernel(const float* __restrict__ agg, const float* __restrict__ b,
                                float* __restrict__ out, int N, int HO) {
    long total = (long)N * HO;
    long stride = (long)gridDim.x * blockDim.x;
    for (long i = (long)blockIdx.x * blockDim.x + threadIdx.x; i < total; i += stride) {
        int j = (int)(i % HO);
        float v = agg[i] + b[j];
        out[i] = (v > 0.f) ? v : (__expf(v) - 1.f);
    }
}

// Layer 3 per-node: h3 = x3 @ W3 (32x2) and its attention terms (H=1, O=2)
__global__ void l3_node_kernel(const float* __restrict__ x, const float* __restrict__ W,
                               const float* __restrict__ att_s, const float* __restrict__ att_d,
                               float* __restrict__ h3, float* __restrict__ as,
                               float* __restrict__ ad, int N) {
    long stride = (long)gridDim.x * blockDim.x;
    for (long n = (long)blockIdx.x * blockDim.x + threadIdx.x; n < N; n += stride) {
        const float* xp = x + (size_t)n * 32;
        float o0 = 0.f, o1 = 0.f;
#pragma unroll
        for (int k = 0; k < 32; ++k) {
            float v = xp[k];
            o0 += v * W[k * 2 + 0];
            o1 += v * W[k * 2 + 1];
        }
        h3[n * 2 + 0] = o0; h3[n * 2 + 1] = o1;
        as[n] = o0 * att_s[0] + o1 * att_s[1];
        ad[n] = o0 * att_d[0] + o1 * att_d[1];
    }
}

// out = log_softmax(agg + b, axis=1) over 2 classes
__global__ void l3_final_kernel(const float* __restrict__ agg, const float* __restrict__ b,
                                float* __restrict__ out, int N) {
    long stride = (long)gridDim.x * blockDim.x;
    for (long n = (long)blockIdx.x * blockDim.x + threadIdx.x; n < N; n += stride) {
        float v0 = agg[n * 2 + 0] + b[0];
        float v1 = agg[n * 2 + 1] + b[1];
        float m = fmaxf(v0, v1);
        float lse = m + __logf(__expf(v0 - m) + __expf(v1 - m));
        out[n * 2 + 0] = v0 - lse;
        out[n * 2 + 1] = v1 - lse;
    }
}

extern "C" void kernel_launch(void* const* d_in, const int* in_sizes, int n_in,
                              void* d_out, int out_size, void* d_ws, size_t ws_size,
                              hipStream_t stream) {
    const float* x   = (const float*)d_in[0];
    const int*   ei  = (const int*)d_in[1];
    const float* W1  = (const float*)d_in[2];
    const float* s1  = (const float*)d_in[3];
    const float* dd1 = (const float*)d_in[4];
    const float* b1  = (const float*)d_in[5];
    const float* W2  = (const float*)d_in[6];
    const float* s2  = (const float*)d_in[7];
    const float* dd2 = (const float*)d_in[8];
    const float* b2  = (const float*)d_in[9];
    const float* W3  = (const float*)d_in[10];
    const float* s3  = (const float*)d_in[11];
    const float* dd3 = (const float*)d_in[12];
    const float* b3  = (const float*)d_in[13];

    const int N = in_sizes[0] / 128;    // FIN = 128
    const int E = in_sizes[1] / 2;
    const int* src = ei;
    const int* dst = ei + E;

    // ---- workspace layout (reused across layers) ----
    char* base = (char*)d_ws;
    size_t off = 0;
    auto take = [&](size_t bytes) -> float* {
        float* r = (float*)(base + off);
        off = (off + bytes + 255) & ~(size_t)255;
        return r;
    };
    float* bufA = take((size_t)N * 128 * sizeof(float)); // h1; later h2|agg2
    float* bufB = take((size_t)N * 128 * sizeof(float)); // agg1->x2; later h3|agg3
    float* as   = take((size_t)N * 4 * sizeof(float));
    float* ad   = take((size_t)N * 4 * sizeof(float));
    float* den  = take((size_t)N * 4 * sizeof(float));
    (void)ws_size; (void)n_in; (void)out_size;

    const int rowTiles = (N + 15) / 16;
    const dim3 TPB(256);
    const dim3 NODE_GRID(1024);
    const dim3 EDGE_GRID(4096);

    // ================= Layer 1: 128 -> (H=4, O=32) concat = 128 =============
    float* h1   = bufA;
    float* agg1 = bufB;
    gemm_wmma_kernel<128, 4><<<dim3(rowTiles, 2), 128, 0, stream>>>(x, W1, h1, N, 128);
    att_coeff_kernel<4, 32><<<NODE_GRID, TPB, 0, stream>>>(h1, s1, dd1, as, ad, N);
    zero_kernel<<<NODE_GRID, TPB, 0, stream>>>(den, (long)N * 4);
    zero_kernel<<<NODE_GRID, TPB, 0, stream>>>(agg1, (long)N * 128);
    edge_denom_kernel<4><<<EDGE_GRID, TPB, 0, stream>>>(src, dst, E, N, as, ad, den);
    edge_aggregate_kernel<4, 32><<<EDGE_GRID, TPB, 0, stream>>>(src, dst, E, N, as, ad, den, h1, agg1);
    bias_elu_kernel<<<NODE_GRID, TPB, 0, stream>>>(agg1, b1, agg1, N, 128);

    // ================= Layer 2: 128 -> (H=2, O=16) concat = 32 ==============
    float* x2   = agg1;                 // bufB
    float* h2   = bufA;                 // N*32 floats
    float* agg2 = bufA + (size_t)N * 32;
    gemm_wmma_kernel<128, 2><<<dim3(rowTiles, 1), 64, 0, stream>>>(x2, W2, h2, N, 32);
    att_coeff_kernel<2, 16><<<NODE_GRID, TPB, 0, stream>>>(h2, s2, dd2, as, ad, N);
    zero_kernel<<<NODE_GRID, TPB, 0, stream>>>(den, (long)N * 2);
    zero_kernel<<<NODE_GRID, TPB, 0, stream>>>(agg2, (long)N * 32);
    edge_denom_kernel<2><<<EDGE_GRID, TPB, 0, stream>>>(src, dst, E, N, as, ad, den);
    edge_aggregate_kernel<2, 16><<<EDGE_GRID, TPB, 0, stream>>>(src, dst, E, N, as, ad, den, h2, agg2);
    bias_elu_kernel<<<NODE_GRID, TPB, 0, stream>>>(agg2, b2, agg2, N, 32);

    // ================= Layer 3: 32 -> (H=1, O=2), mean + log_softmax ========
    float* x3   = agg2;
    float* h3   = bufB;                 // N*2 floats
    float* agg3 = bufB + (size_t)N * 2;
    l3_node_kernel<<<NODE_GRID, TPB, 0, stream>>>(x3, W3, s3, dd3, h3, as, ad, N);
    zero_kernel<<<NODE_GRID, TPB, 0, stream>>>(den, (long)N);
    zero_kernel<<<NODE_GRID, TPB, 0, stream>>>(agg3, (long)N * 2);
    edge_denom_kernel<1><<<EDGE_GRID, TPB, 0, stream>>>(src, dst, E, N, as, ad, den);
    edge_aggregate_kernel<1, 2><<<EDGE_GRID, TPB, 0, stream>>>(src, dst, E, N, as, ad, den, h3, agg3);
    l3_final_kernel<<<NODE_GRID, TPB, 0, stream>>>(agg3, b3, (float*)d_out, N);
}